// CarryII_36472862278061
// MI455X (gfx1250) — compile-verified
//
#include <hip/hip_runtime.h>

// ---------------------------------------------------------------------------
// CarryII: 7x (gather(src_idx) -> segment_sum(dst_idx)) over D=32 f32 features.
// Pure memory-bound scatter-add: zero-FLOP, so no WMMA path applies.
// MI455X strategy:
//   * 8 lanes per edge, b128 loads  -> wave32 covers 4 contiguous 128B rows
//   * global_atomic_add_f32 (no-return) for segment accumulation
//   * global_prefetch_b8 of next iteration's gathered row (gfx1250 prefetch)
//   * non-temporal b128 stores for the 1.66GB output zero-init (don't thrash
//     the 192MB L2 that we want holding u2 [128MB, fully resident] / u3)
// ---------------------------------------------------------------------------

typedef float v4f __attribute__((ext_vector_type(4)));

#define THREADS 256
#define MAX_BLOCKS 12288LL   // cap grid; grid-stride loop gives the prefetch a pipeline

__global__ __launch_bounds__(THREADS) void zero_fill_nt(v4f* __restrict__ out,
                                                        long long n4) {
  const long long stride = (long long)gridDim.x * blockDim.x;
  for (long long i = (long long)blockIdx.x * blockDim.x + threadIdx.x; i < n4;
       i += stride) {
    v4f z = (v4f)0.0f;
    __builtin_nontemporal_store(z, out + i);   // global_store_b128 TH=NT
  }
}

// One relation: out[dst[e]*32 + c*4 .. +3] += src[sidx[e]*32 + c*4 .. +3]
// t indexes (edge, chunk) pairs: e = t>>3, c = t&7  (8 float4 chunks per row)
__global__ __launch_bounds__(THREADS) void gather_scatter_add(
    const v4f* __restrict__ src,        // (n_src, 8) rows of float4
    const int* __restrict__ sidx,
    const int* __restrict__ didx,
    float* __restrict__ out,            // base of this relation's output
    int n_edges) {
  const long long total = (long long)n_edges * 8;
  const long long stride = (long long)gridDim.x * blockDim.x;
  for (long long t = (long long)blockIdx.x * blockDim.x + threadIdx.x;
       t < total; t += stride) {
    const int e = (int)(t >> 3);
    const int c = (int)(t & 7);

    // Prefetch next iteration's gathered chunk (emits global_prefetch_b8).
    const long long tn = t + stride;
    if (tn < total) {
      const int en = (int)(tn >> 3);
      const int cn = (int)(tn & 7);
      __builtin_prefetch((const void*)(src + (size_t)sidx[en] * 8 + cn), 0, 1);
    }

    const int si = sidx[e];
    const int di = didx[e];
    const v4f v = src[(size_t)si * 8 + c];          // global_load_b128
    float* o = out + (size_t)di * 32 + (size_t)c * 4;
    atomicAdd(o + 0, v.x);                          // global_atomic_add_f32
    atomicAdd(o + 1, v.y);
    atomicAdd(o + 2, v.z);
    atomicAdd(o + 3, v.w);
  }
}

static inline int grid_for(long long work_items) {
  long long b = (work_items + THREADS - 1) / THREADS;
  if (b > MAX_BLOCKS) b = MAX_BLOCKS;
  if (b < 1) b = 1;
  return (int)b;
}

extern "C" void kernel_launch(void* const* d_in, const int* in_sizes, int n_in,
                              void* d_out, int out_size, void* d_ws,
                              size_t ws_size, hipStream_t stream) {
  (void)n_in; (void)d_ws; (void)ws_size;

  const float* u2 = (const float*)d_in[0];
  const float* u3 = (const float*)d_in[1];

  const int N3 = in_sizes[2];   // length of s_b0a
  const int N4 = in_sizes[6];   // length of s_b0t
  const int D = 32;

  float* out = (float*)d_out;

  // Output is poisoned by the harness; segment-sum needs zeroed accumulators.
  const long long n4 = (long long)out_size / 4;
  zero_fill_nt<<<grid_for(n4), THREADS, 0, stream>>>((v4f*)out, n4);

  // Relation table: {source features, src-idx input, dst-idx input, out offset, n_edges}
  struct Rel {
    const float* src;
    int s_slot, d_slot;
    size_t off;
    int n;
  };
  const size_t sN3 = (size_t)N3 * D;
  const size_t sN4 = (size_t)N4 * D;
  const Rel rels[7] = {
      {u2,  2,  3, 0,                 N3},  // u_left
      {u2,  4,  5, sN3,               N3},  // u_right
      {u2,  6,  7, 2 * sN3,           N4},  // u_bond_left
      {u2,  8,  9, 2 * sN3 + sN4,     N4},  // u_bond_center
      {u2, 10, 11, 2 * sN3 + 2 * sN4, N4},  // u_bond_right
      {u3, 12, 13, 2 * sN3 + 3 * sN4, N4},  // u_angle_left
      {u3, 14, 15, 2 * sN3 + 4 * sN4, N4},  // u_angle_right
  };

  for (int r = 0; r < 7; ++r) {
    const Rel& R = rels[r];
    if (R.n <= 0) continue;
    gather_scatter_add<<<grid_for((long long)R.n * 8), THREADS, 0, stream>>>(
        (const v4f*)R.src, (const int*)d_in[R.s_slot],
        (const int*)d_in[R.d_slot], out + R.off, R.n);
  }
}